// CVHI_64020782514291
// MI455X (gfx1250) — compile-verified
//
#include <hip/hip_runtime.h>

// ---------------------------------------------------------------------------
// Fused CDNA5 (gfx1250) forward for the sparse-GAT recurrent model.
// One workgroup (256 threads = 8 x wave32) processes PW=8 (b,t) positions
// end-to-end with all node features resident in LDS. Dense d x d linears run
// on V_WMMA_F32_16X16X4_F32 with fully branchless, padded tiles:
//   - A fragments load unconditionally from 48-row LDS buffers (padding rows
//     only affect D rows that are never consumed),
//   - B fragments load from clamped (in-bounds) global addresses (duplicated
//     columns only affect D columns that land in LDS padding),
//   - all GEMM operand loads are 8-byte (ds_load_2addr_b64 / global_load_b64),
//   - node-embedding adds are compile-time specialized (no branches, literal
//     modulo).
// Attention top-3/softmax, layernorm and head MLPs run on VALU.
// ---------------------------------------------------------------------------

typedef float v2f __attribute__((ext_vector_type(2)));
typedef float v8f __attribute__((ext_vector_type(8)));

#define PW 8      // positions per workgroup
#define LD 48     // LDS row stride in floats (== padded row count)
#define NW 8      // waves per workgroup

struct Params {
  const float* state;                                   // (B*T, 6)
  const float* A_vis;                                   // (5,5)
  const float* alpha_raw;                               // (1)
  const float* b_h2v;                                   // (5)
  const float* c_h2v;                                   // (5)
  const float* hgat_kW[2]; const float* hgat_kb[2];
  const float* hgat_oW[2]; const float* hgat_ob[2];
  const float* hgat_qW[2]; const float* hgat_qb[2];
  const float* hgat_vW[2]; const float* hgat_vb[2];
  const float* hhead1W; const float* hhead1b;           // (24,48),(24)
  const float* hhead2W; const float* hhead2b;           // (1,24),(1)
  const float* hid_emb;                                 // (6,48)
  const float* hnorm_b[2]; const float* hnorm_g[2];
  const float* hproj1W; const float* hproj1b;           // (48,2),(48)
  const float* hproj2W; const float* hproj2b;           // (48,48),(48)
  const float* r_hidden;                                // (1)
  const float* r_vis;                                   // (5)
  const float* vgat_kW[2]; const float* vgat_kb[2];
  const float* vgat_oW[2]; const float* vgat_ob[2];
  const float* vgat_qW[2]; const float* vgat_qb[2];
  const float* vgat_vW[2]; const float* vgat_vb[2];
  const float* vhead1W; const float* vhead1b;           // (12,24),(12)
  const float* vhead2W; const float* vhead2b;           // (1,12),(1)
  const float* vis_emb;                                 // (5,24)
  const float* vnorm_b[2]; const float* vnorm_g[2];
  const float* vproj1W; const float* vproj1b;           // (24,2),(24)
  const float* vproj2W; const float* vproj2b;           // (24,24),(24)
};

__device__ __forceinline__ float gelu_f(float x) {
  return 0.5f * x * (1.0f + erff(x * 0.70710678118654752440f));
}

__device__ __forceinline__ float dot24(const float* a, const float* b) {
  const float4* a4 = (const float4*)a;
  const float4* b4 = (const float4*)b;
  float s = 0.f;
#pragma unroll
  for (int u = 0; u < 6; ++u) {
    const float4 x = a4[u], y = b4[u];
    s = fmaf(x.x, y.x, s); s = fmaf(x.y, y.y, s);
    s = fmaf(x.z, y.z, s); s = fmaf(x.w, y.w, s);
  }
  return s;
}

// to[0:24) = e0*v0 + e1*v1 + e2*v2, all 16B-aligned LDS rows.
__device__ __forceinline__ void wsum24(float* to, const float* v0, const float* v1,
                                       const float* v2, float e0, float e1, float e2) {
  const float4* a = (const float4*)v0;
  const float4* b = (const float4*)v1;
  const float4* c = (const float4*)v2;
  float4* o = (float4*)to;
#pragma unroll
  for (int u = 0; u < 6; ++u) {
    const float4 x = a[u], y = b[u], z = c[u];
    float4 r;
    r.x = fmaf(e0, x.x, fmaf(e1, y.x, e2 * z.x));
    r.y = fmaf(e0, x.y, fmaf(e1, y.y, e2 * z.y));
    r.z = fmaf(e0, x.z, fmaf(e1, y.z, e2 * z.z));
    r.w = fmaf(e0, x.w, fmaf(e1, y.w, e2 * z.w));
    o[u] = r;
  }
}

// Up to NMAT GEMMs sharing the A operand:
//   Yi[48 x DOUT] = X[48 x K] @ Wi^T + Bi (+ emb[(row % NODEMOD) * ES + n])
// X, Yi in LDS (row stride LD = 48 == padded row count); Wi row-major
// (DOUT x K) in global memory (WGP$/L2-hot, uniform per 16-lane group).
// All loads/stores branchless: A rows always in-bounds (padding rows feed only
// unused D rows), B/bias/emb column index clamped to DOUT-1 (duplicated
// columns feed only LDS padding columns of Y).
template <int K, int DOUT, int NMAT, bool EMB, int NODEMOD, int ES>
__device__ __forceinline__ void gemmT(
    const float* X,
    const float* W0, const float* B0, float* Y0,
    const float* W1, const float* B1, float* Y1,
    const float* W2, const float* B2, float* Y2,
    const float* emb, int wave, int lane)
{
  constexpr int colTiles = (DOUT + 15) / 16;
  constexpr int tilesPer = 3 * colTiles;        // rowTiles == 3 (48 rows)
  const int hi = lane >> 4;                     // 0 or 1
  const int lm = lane & 15;
  for (int job = wave; job < NMAT * tilesPer; job += NW) {
    const int mat = job / tilesPer;
    const int rem = job - mat * tilesPer;
    const int rt  = rem / colTiles;
    const int ct  = rem - rt * colTiles;
    const int m  = rt * 16 + lm;
    const int n  = ct * 16 + lm;
    const int nc = (n < DOUT) ? n : (DOUT - 1);   // branchless clamp
    const float* W = (mat == 0) ? W0 : ((mat == 1) ? W1 : W2);
    const float* B = (mat == 0) ? B0 : ((mat == 1) ? B1 : B2);
    float*       Y = (mat == 0) ? Y0 : ((mat == 1) ? Y1 : Y2);
    v8f acc = {0.f, 0.f, 0.f, 0.f, 0.f, 0.f, 0.f, 0.f};
#pragma unroll
    for (int k0 = 0; k0 < K; k0 += 4) {
      const int ka = k0 + 2 * hi;
      const v2f a = *(const v2f*)(X + m * LD + ka);   // ds_load_b64 (paired)
      const v2f b = *(const v2f*)(W + nc * K + ka);   // global_load_b64
      acc = __builtin_amdgcn_wmma_f32_16x16x4_f32(
          false, a, false, b, (short)0, acc, false, false);
    }
    const float bn = B[nc];
#pragma unroll
    for (int j = 0; j < 8; ++j) {
      const int mm = rt * 16 + 8 * hi + j;
      float val = acc[j] + bn;
      if (EMB) val += emb[(mm % NODEMOD) * ES + nc];  // literal modulo, no branch
      Y[mm * LD + n] = val;                           // ds_store_b32
    }
  }
}

__global__ __launch_bounds__(256)
void fused_gat_forward(Params Pp, float* __restrict__ out, int Pn) {
  __shared__ float xs[LD * LD];   // node features (residual stream)
  __shared__ float qs[LD * LD];   // q / o-proj output / head partials
  __shared__ float ks[LD * LD];
  __shared__ float vs[LD * LD];
  __shared__ float ts[LD * LD];   // pre-proj activations / attention out
  __shared__ float ss[PW * 6];    // raw state values for this tile
  __shared__ float ls[PW * 6];    // log(clip(state))

  const int tid  = threadIdx.x;
  const int lane = tid & 31;
  const int wave = tid >> 5;
  const int p0   = blockIdx.x * PW;
  if (p0 >= Pn) return;

  // Output layout (flat fp32): main (Pn,6) | attnV0 (Pn,1,5,5) | attnV1 |
  //                            attnH0 (Pn,2,6,6) | attnH1
  const size_t aV[2] = { (size_t)Pn * 6,
                         (size_t)Pn * 6 + (size_t)Pn * 25 };
  const size_t aH[2] = { (size_t)Pn * 6 + (size_t)Pn * 50,
                         (size_t)Pn * 6 + (size_t)Pn * 50 + (size_t)Pn * 72 };

  if (tid < PW * 6) {                     // coalesced state preload + logs
    const float x = Pp.state[(size_t)p0 * 6 + tid];
    ss[tid] = x;
    ls[tid] = logf(fmaxf(x, 1e-6f));
  }
  __syncthreads();

  // ===================== visible pipeline (5 nodes, d=24) =====================
  for (int e = tid; e < PW * 5 * 24; e += 256) {   // t1 = gelu(vproj1 @ feats)
    const int r = e / 24, c = e % 24;
    const int si = (r / 5) * 6 + (r % 5);
    const float t = Pp.vproj1W[c * 2 + 0] * ss[si]
                  + Pp.vproj1W[c * 2 + 1] * ls[si] + Pp.vproj1b[c];
    ts[r * LD + c] = gelu_f(t);
  }
  __syncthreads();
  gemmT<24, 24, 1, true, 5, 24>(ts, Pp.vproj2W, Pp.vproj2b, xs,
                                nullptr, nullptr, nullptr,
                                nullptr, nullptr, nullptr,
                                Pp.vis_emb, wave, lane);
  __syncthreads();

#pragma unroll
  for (int layer = 0; layer < 2; ++layer) {
    gemmT<24, 24, 3, false, 1, 0>(xs,
          Pp.vgat_qW[layer], Pp.vgat_qb[layer], qs,
          Pp.vgat_kW[layer], Pp.vgat_kb[layer], ks,
          Pp.vgat_vW[layer], Pp.vgat_vb[layer], vs,
          nullptr, wave, lane);
    __syncthreads();

    if (tid < PW * 5) {                 // 1 head, N=5, dh=24, top-3
      const int pl = tid / 5, i = tid % 5, rb = pl * 5;
      const float* qi = &qs[(rb + i) * LD];
      float sc[5];
      for (int j = 0; j < 5; ++j)
        sc[j] = dot24(qi, &ks[(rb + j) * LD]) * 0.20412414523193150818f;
      int i0 = 0;
      for (int j = 1; j < 5; ++j) if (sc[j] > sc[i0]) i0 = j;
      int i1 = -1;
      for (int j = 0; j < 5; ++j) if (j != i0 && (i1 < 0 || sc[j] > sc[i1])) i1 = j;
      int i2 = -1;
      for (int j = 0; j < 5; ++j)
        if (j != i0 && j != i1 && (i2 < 0 || sc[j] > sc[i2])) i2 = j;
      const float mx = sc[i0];
      float e0 = expf(sc[i0] - mx), e1 = expf(sc[i1] - mx), e2 = expf(sc[i2] - mx);
      const float inv = 1.f / (e0 + e1 + e2);
      e0 *= inv; e1 *= inv; e2 *= inv;
      wsum24(&ts[(rb + i) * LD], &vs[(rb + i0) * LD], &vs[(rb + i1) * LD],
             &vs[(rb + i2) * LD], e0, e1, e2);
      float ar[5] = {0.f, 0.f, 0.f, 0.f, 0.f};
      ar[i0] = e0; ar[i1] = e1; ar[i2] = e2;
      float* ap = out + aV[layer] + (size_t)(p0 + pl) * 25 + (size_t)i * 5;
      for (int j = 0; j < 5; ++j) ap[j] = ar[j];
    }
    __syncthreads();

    gemmT<24, 24, 1, false, 1, 0>(ts, Pp.vgat_oW[layer], Pp.vgat_ob[layer], qs,
                                  nullptr, nullptr, nullptr,
                                  nullptr, nullptr, nullptr,
                                  nullptr, wave, lane);
    __syncthreads();

    if (tid < PW * 5) {                 // residual + layernorm (d=24)
      float* xr = &xs[tid * LD];
      const float* dr = &qs[tid * LD];
      float h[24];
      float mean = 0.f;
      for (int c = 0; c < 24; ++c) { h[c] = xr[c] + dr[c]; mean += h[c]; }
      mean *= (1.f / 24.f);
      float var = 0.f;
      for (int c = 0; c < 24; ++c) { const float d = h[c] - mean; var += d * d; }
      var *= (1.f / 24.f);
      const float inv = rsqrtf(var + 1e-5f);
      const float* g = Pp.vnorm_g[layer];
      const float* b = Pp.vnorm_b[layer];
      for (int c = 0; c < 24; ++c) xr[c] = (h[c] - mean) * inv * g[c] + b[c];
    }
    __syncthreads();
  }

  // visible head MLP 24->12: one (row, unit) partial per thread into qs.
  for (int e = tid; e < PW * 5 * 12; e += 256) {
    const int r = e / 12, u = e % 12;
    float t = Pp.vhead1b[u];
    const float* xr = &xs[r * LD];
    for (int c = 0; c < 24; ++c) t = fmaf(Pp.vhead1W[u * 24 + c], xr[c], t);
    qs[r * 12 + u] = Pp.vhead2W[u] * gelu_f(t);
  }
  __syncthreads();

  // visible_lr assembly (reads qs partials) + hidden feature stack (writes ts)
  if (tid < PW * 5) {
    const int pl = tid / 5, i = tid % 5;
    float s = Pp.vhead2b[0];
    for (int u = 0; u < 12; ++u) s += qs[tid * 12 + u];
    const float alpha = 1.f / (1.f + expf(-Pp.alpha_raw[0]));
    float acc = Pp.r_vis[i];
    for (int j = 0; j < 5; ++j)
      acc = fmaf(Pp.A_vis[i * 5 + j], ss[pl * 6 + j], acc);
    const float ht = ss[pl * 6 + 5];
    acc += alpha * s + ht * Pp.b_h2v[i] + ht * ht * Pp.c_h2v[i];
    out[(size_t)(p0 + pl) * 6 + i] = acc;
  }

  // ===================== hidden pipeline (6 nodes, d=48) =====================
  for (int e = tid; e < PW * 6 * 48; e += 256) {
    const int r = e / 48, c = e % 48;
    const float t = Pp.hproj1W[c * 2 + 0] * ss[r]
                  + Pp.hproj1W[c * 2 + 1] * ls[r] + Pp.hproj1b[c];
    ts[r * LD + c] = gelu_f(t);
  }
  __syncthreads();
  gemmT<48, 48, 1, true, 6, 48>(ts, Pp.hproj2W, Pp.hproj2b, xs,
                                nullptr, nullptr, nullptr,
                                nullptr, nullptr, nullptr,
                                Pp.hid_emb, wave, lane);
  __syncthreads();

#pragma unroll
  for (int layer = 0; layer < 2; ++layer) {
    gemmT<48, 48, 3, false, 1, 0>(xs,
          Pp.hgat_qW[layer], Pp.hgat_qb[layer], qs,
          Pp.hgat_kW[layer], Pp.hgat_kb[layer], ks,
          Pp.hgat_vW[layer], Pp.hgat_vb[layer], vs,
          nullptr, wave, lane);
    __syncthreads();

    if (tid < PW * 12) {                // 2 heads, N=6, dh=24, top-3
      const int pl = tid / 12;
      const int rem = tid % 12;
      const int head = rem / 6, i = rem % 6, rb = pl * 6;
      const float* qi = &qs[(rb + i) * LD + head * 24];
      float sc[6];
      for (int j = 0; j < 6; ++j)
        sc[j] = dot24(qi, &ks[(rb + j) * LD + head * 24]) * 0.20412414523193150818f;
      int i0 = 0;
      for (int j = 1; j < 6; ++j) if (sc[j] > sc[i0]) i0 = j;
      int i1 = -1;
      for (int j = 0; j < 6; ++j) if (j != i0 && (i1 < 0 || sc[j] > sc[i1])) i1 = j;
      int i2 = -1;
      for (int j = 0; j < 6; ++j)
        if (j != i0 && j != i1 && (i2 < 0 || sc[j] > sc[i2])) i2 = j;
      const float mx = sc[i0];
      float e0 = expf(sc[i0] - mx), e1 = expf(sc[i1] - mx), e2 = expf(sc[i2] - mx);
      const float inv = 1.f / (e0 + e1 + e2);
      e0 *= inv; e1 *= inv; e2 *= inv;
      const int ho = head * 24;
      wsum24(&ts[(rb + i) * LD + ho], &vs[(rb + i0) * LD + ho],
             &vs[(rb + i1) * LD + ho], &vs[(rb + i2) * LD + ho], e0, e1, e2);
      float ar[6] = {0.f, 0.f, 0.f, 0.f, 0.f, 0.f};
      ar[i0] = e0; ar[i1] = e1; ar[i2] = e2;
      float* ap = out + aH[layer]
                + ((size_t)(p0 + pl) * 2 + head) * 36 + (size_t)i * 6;
      for (int j = 0; j < 6; ++j) ap[j] = ar[j];
    }
    __syncthreads();

    gemmT<48, 48, 1, false, 1, 0>(ts, Pp.hgat_oW[layer], Pp.hgat_ob[layer], qs,
                                  nullptr, nullptr, nullptr,
                                  nullptr, nullptr, nullptr,
                                  nullptr, wave, lane);
    __syncthreads();

    if (tid < PW * 6) {                 // residual + layernorm (d=48)
      float* xr = &xs[tid * LD];
      const float* dr = &qs[tid * LD];
      float h[48];
      float mean = 0.f;
      for (int c = 0; c < 48; ++c) { h[c] = xr[c] + dr[c]; mean += h[c]; }
      mean *= (1.f / 48.f);
      float var = 0.f;
      for (int c = 0; c < 48; ++c) { const float d = h[c] - mean; var += d * d; }
      var *= (1.f / 48.f);
      const float inv = rsqrtf(var + 1e-5f);
      const float* g = Pp.hnorm_g[layer];
      const float* b = Pp.hnorm_b[layer];
      for (int c = 0; c < 48; ++c) xr[c] = (h[c] - mean) * inv * g[c] + b[c];
    }
    __syncthreads();
  }

  // hidden head MLP 48->24: (position, unit) partials into ts, then reduce.
  if (tid < PW * 24) {
    const int pl = tid / 24, u = tid % 24;
    const float* xr = &xs[(pl * 6 + 5) * LD];
    float t = Pp.hhead1b[u];
    for (int c = 0; c < 48; ++c) t = fmaf(Pp.hhead1W[u * 48 + c], xr[c], t);
    ts[pl * 24 + u] = Pp.hhead2W[u] * gelu_f(t);
  }
  __syncthreads();
  if (tid < PW) {
    float s = Pp.hhead2b[0] + Pp.r_hidden[0];
    for (int u = 0; u < 24; ++u) s += ts[tid * 24 + u];
    out[(size_t)(p0 + tid) * 6 + 5] = s;
  }
}

extern "C" void kernel_launch(void* const* d_in, const int* in_sizes, int n_in,
                              void* d_out, int out_size, void* d_ws, size_t ws_size,
                              hipStream_t stream) {
  (void)n_in; (void)out_size; (void)d_ws; (void)ws_size;
  const float* const* F = (const float* const*)d_in;
  Params p;
  // Input order: state first (setup_inputs dict order), then params leaves in
  // jax pytree order (dict keys sorted; lin={'W','b'} -> W,b; ln={'g','b'} -> b,g;
  // gat dict {'q','k','v','o'} -> k,o,q,v).
  p.state = F[0];
  p.A_vis = F[1]; p.alpha_raw = F[2]; p.b_h2v = F[3]; p.c_h2v = F[4];
  int i = 5;
  for (int l = 0; l < 2; ++l) {
    p.hgat_kW[l] = F[i++]; p.hgat_kb[l] = F[i++];
    p.hgat_oW[l] = F[i++]; p.hgat_ob[l] = F[i++];
    p.hgat_qW[l] = F[i++]; p.hgat_qb[l] = F[i++];
    p.hgat_vW[l] = F[i++]; p.hgat_vb[l] = F[i++];
  }
  p.hhead1W = F[i++]; p.hhead1b = F[i++];
  p.hhead2W = F[i++]; p.hhead2b = F[i++];
  p.hid_emb = F[i++];
  for (int l = 0; l < 2; ++l) { p.hnorm_b[l] = F[i++]; p.hnorm_g[l] = F[i++]; }
  p.hproj1W = F[i++]; p.hproj1b = F[i++];
  p.hproj2W = F[i++]; p.hproj2b = F[i++];
  p.r_hidden = F[i++]; p.r_vis = F[i++];
  for (int l = 0; l < 2; ++l) {
    p.vgat_kW[l] = F[i++]; p.vgat_kb[l] = F[i++];
    p.vgat_oW[l] = F[i++]; p.vgat_ob[l] = F[i++];
    p.vgat_qW[l] = F[i++]; p.vgat_qb[l] = F[i++];
    p.vgat_vW[l] = F[i++]; p.vgat_vb[l] = F[i++];
  }
  p.vhead1W = F[i++]; p.vhead1b = F[i++];
  p.vhead2W = F[i++]; p.vhead2b = F[i++];
  p.vis_emb = F[i++];
  for (int l = 0; l < 2; ++l) { p.vnorm_b[l] = F[i++]; p.vnorm_g[l] = F[i++]; }
  p.vproj1W = F[i++]; p.vproj1b = F[i++];
  p.vproj2W = F[i++]; p.vproj2b = F[i++];

  const int Pn = in_sizes[0] / 6;       // B*T positions
  dim3 grid((Pn + PW - 1) / PW), block(256);
  hipLaunchKernelGGL(fused_gat_forward, grid, block, 0, stream,
                     p, (float*)d_out, Pn);
}